// CTCLayer_63213328662988
// MI455X (gfx1250) — compile-verified
//
#include <hip/hip_runtime.h>
#include <cstdint>

#define EPSF   1e-7f
#define NEGV  (-1e30f)

typedef float v2f __attribute__((ext_vector_type(2)));
typedef float v8f __attribute__((ext_vector_type(8)));

// ---- CDNA5 async global->LDS copy (16B per lane), inline asm (portable across toolchains) ----
static __device__ __forceinline__ void async_b128(const float* gsrc, float* ldst) {
    unsigned l = (unsigned)(uintptr_t)ldst;   // low 32 bits of generic shared ptr == LDS offset
    asm volatile("global_load_async_to_lds_b128 %0, %1, off"
                 :: "v"(l), "v"(gsrc) : "memory");
}

template <int N>
static __device__ __forceinline__ void wait_asynccnt() {
    asm volatile("s_wait_asynccnt %0" :: "n"(N) : "memory");
}

__global__ __launch_bounds__(256)
void ctc_loss_kernel(const int* __restrict__ labels,   // (B, L) int32
                     const float* __restrict__ y_pred, // (B, T, C) f32
                     float* __restrict__ out)          // (B,) f32
{
    constexpr int T = 512, C = 512, L = 64, S = 2 * L + 1;  // S = 129
    constexpr int RS = 516;   // padded LDS row stride (floats): 516 mod 64 banks -> conflict-free

    __shared__ __align__(16) float tile[2][16 * RS];  // 66 KB: double-buffered 16-row tiles
    __shared__ __align__(16) float rbuf[2][C];        // 4 KB : double-buffered scan rows
    __shared__ float logZ[T];
    __shared__ float partial[8][16];
    __shared__ int   ext[S + 3];
    __shared__ int   skipv[S + 3];
    __shared__ float alpha[S + 3];

    const int tid = threadIdx.x;
    const int b   = blockIdx.x;
    const float* yb = y_pred + (size_t)b * T * C;

    // ---- extended label sequence + skip mask ----
    if (tid < S) ext[tid] = (tid & 1) ? labels[b * L + (tid >> 1)] : (C - 1);
    __syncthreads();
    if (tid < S) skipv[tid] = (tid >= 2) && (ext[tid] != C - 1) && (ext[tid] != ext[tid - 2]);

    // ================= Phase 1: row sums via async tiles + f32 WMMA =================
    // tile k holds rows t = 16k..16k+15 ; 2048 segments of 16B -> 8 async issues/thread
    auto issue_tile = [&](int k) {
        const float* gb = yb + (size_t)(k * 16) * C;
        float* lb = &tile[k & 1][0];
        #pragma unroll
        for (int i = 0; i < 8; ++i) {
            int idx = tid + i * 256;      // 0..2047
            int row = idx >> 7;           // 16 rows
            int seg = idx & 127;          // 128 x 16B per row
            async_b128(gb + row * C + seg * 4, lb + row * RS + seg * 4);
        }
    };

    issue_tile(0);
    wait_asynccnt<0>();
    __syncthreads();

    const int wave = tid >> 5;
    const int lane = tid & 31;
    const int tl   = lane & 15;
    const int g2   = (lane >> 4) * 2;     // K offset 0 or 2 for f32 A-matrix layout

    for (int k = 0; k < 32; ++k) {
        if (k + 1 < 32) issue_tile(k + 1);          // overlap loads with WMMA compute

        const float* tp = &tile[k & 1][0];
        v8f acc = {0.f, 0.f, 0.f, 0.f, 0.f, 0.f, 0.f, 0.f};
        v2f bones; bones.x = 1.f; bones.y = 1.f;    // B = ones(4x16) -> D[i][j] = sum_k A[i][k]
        // wave w reduces classes [w*64, w*64+64) in 16 K-chunks of 4
        #pragma unroll
        for (int kk = 0; kk < 16; ++kk) {
            int cb = (wave * 16 + kk) * 4 + g2;
            v2f a;
            a.x = tp[tl * RS + cb];        // A[M=tl][K=g2]
            a.y = tp[tl * RS + cb + 1];    // A[M=tl][K=g2+1]
            acc = __builtin_amdgcn_wmma_f32_16x16x4_f32(
                      false, a, false, bones, (short)0, acc, false, false);
        }
        // D: vgpr r, lanes 0-15 -> M=r ; lanes 16-31 -> M=8+r (replicated over N)
        if (tl == 0) {
            int half = (lane >> 4) * 8;
            #pragma unroll
            for (int r = 0; r < 8; ++r) partial[wave][half + r] = acc[r];
        }
        __syncthreads();
        if (tid < 16) {
            float s = 0.f;
            #pragma unroll
            for (int w = 0; w < 8; ++w) s += partial[w][tid];
            logZ[k * 16 + tid] = __logf(s + (float)C * EPSF);
        }
        wait_asynccnt<0>();   // next tile landed (overlapped with compute above)
        __syncthreads();
    }

    // ================= Phase 2: forward scan with async double-buffered rows =================
    if (tid < 128) {
        async_b128(yb + tid * 4,               &rbuf[0][tid * 4]);   // row 0
        async_b128(yb + (size_t)C + tid * 4,   &rbuf[1][tid * 4]);   // row 1
    }
    wait_asynccnt<1>();       // row 0 landed (in-order completion)
    __syncthreads();

    if (tid < S) {
        float lp = __logf(rbuf[0][ext[tid]] + EPSF) - logZ[0];
        alpha[tid] = (tid < 2) ? lp : NEGV;
    }
    __syncthreads();

    for (int t = 1; t < T; ++t) {
        if (t + 1 < T) {
            if (tid < 128)
                async_b128(yb + (size_t)(t + 1) * C + tid * 4, &rbuf[(t + 1) & 1][tid * 4]);
            wait_asynccnt<1>();   // row t landed; row t+1 still in flight
        } else {
            wait_asynccnt<0>();
        }
        __syncthreads();

        float nv = 0.f;
        if (tid < S) {
            float lp = __logf(rbuf[t & 1][ext[tid]] + EPSF) - logZ[t];
            float a0 = alpha[tid];
            float a1 = (tid >= 1)  ? alpha[tid - 1] : NEGV;
            float a2 = skipv[tid]  ? alpha[tid - 2] : NEGV;
            float m  = fmaxf(a0, fmaxf(a1, a2));
            nv = m + __logf(__expf(a0 - m) + __expf(a1 - m) + __expf(a2 - m)) + lp;
        }
        __syncthreads();
        if (tid < S) alpha[tid] = nv;
    }

    __syncthreads();
    if (tid == 0) {
        float e1 = alpha[S - 1], e2 = alpha[S - 2];
        float m  = fmaxf(e1, e2);
        out[b] = -(m + __logf(__expf(e1 - m) + __expf(e2 - m)));
    }
}

extern "C" void kernel_launch(void* const* d_in, const int* in_sizes, int n_in,
                              void* d_out, int out_size, void* d_ws, size_t ws_size,
                              hipStream_t stream) {
    (void)in_sizes; (void)n_in; (void)d_ws; (void)ws_size; (void)out_size;
    const int*   y_true = (const int*)d_in[0];    // (128, 64) int32
    const float* y_pred = (const float*)d_in[1];  // (128, 512, 512) f32
    float*       out    = (float*)d_out;          // 128 losses
    hipLaunchKernelGGL(ctc_loss_kernel, dim3(128), dim3(256), 0, stream,
                       y_true, y_pred, out);
}